// BoundaryLoss_47124381172366
// MI455X (gfx1250) — compile-verified
//
#include <hip/hip_runtime.h>
#include <math.h>

#define BB 8
#define HH 256
#define WW 256
#define BIGV 1e9f

typedef float v2f __attribute__((ext_vector_type(2)));
typedef float v8f __attribute__((ext_vector_type(8)));

// Async global->LDS staging (ASYNCcnt path), guarded so compile never breaks
// if this toolchain doesn't declare the gfx1250 async builtins.
#if defined(__gfx1250__) && __has_builtin(__builtin_amdgcn_global_load_async_to_lds_b32) && __has_builtin(__builtin_amdgcn_s_wait_asynccnt)
#define USE_ASYNC_LDS 1
typedef __attribute__((address_space(1))) int async_gint;   // global operand type
typedef __attribute__((address_space(3))) int async_lint;   // LDS operand type
#else
#define USE_ASYNC_LDS 0
#endif

// Reduce one fp32 partial across the 32 lanes of a wave using the f32 tensor op:
// D = A(16x4) * Ones(4x16). With A[m][0]=v(lane m), A[m][2]=v(lane m+16) (rest 0),
// every D column holds the 16 row-sums r(m)=v(m)+v(m+16). Summing a lane's 8 D
// VGPRs gives sum(r(0..7)) on lanes 0-15 and sum(r(8..15)) on lanes 16-31; one
// shfl_xor(16) completes the 32-lane total (uniform across the wave).
__device__ __forceinline__ float wave_reduce32(float v) {
    v2f a; a[0] = v;    a[1] = 0.0f;
    v2f b; b[0] = 1.0f; b[1] = 1.0f;
    v8f c = {};
    c = __builtin_amdgcn_wmma_f32_16x16x4_f32(false, a, false, b, (short)0, c, false, false);
    float r = ((c[0] + c[1]) + (c[2] + c[3])) + ((c[4] + c[5]) + (c[6] + c[7]));
    r += __shfl_xor(r, 16, 32);
    return r;
}

// Pass 1: exact 1-D lower envelope along H for both transforms (fg and bg).
// One block = 4 adjacent columns of one image. f[j] = d2_init[j] + j^2 staged in
// LDS; inner loop is a uniform-address (broadcast) ds_load_b128 + add/min chain.
// d2'[i] = i^2 + min_j (f[j] - 2*i*j)  — bit-exact integers on the finite path.
__global__ void edt_h_pass(const int* __restrict__ tg,
                           float* __restrict__ d2p, float* __restrict__ d2n) {
    __shared__ __align__(16) float sP[HH * 4];
    __shared__ __align__(16) float sN[HH * 4];
    const int b  = blockIdx.x >> 6;          // 64 blocks per image
    const int w0 = (blockIdx.x & 63) << 2;   // 4 columns per block
    const int i  = threadIdx.x;              // row index (0..255)
    const float i2 = (float)(i * i);

    const int4 t4 = *(const int4*)(tg + (b * HH + i) * WW + w0);
    sP[i * 4 + 0] = (t4.x == 1 ? 0.0f : BIGV) + i2;
    sP[i * 4 + 1] = (t4.y == 1 ? 0.0f : BIGV) + i2;
    sP[i * 4 + 2] = (t4.z == 1 ? 0.0f : BIGV) + i2;
    sP[i * 4 + 3] = (t4.w == 1 ? 0.0f : BIGV) + i2;
    sN[i * 4 + 0] = (t4.x == 1 ? BIGV : 0.0f) + i2;
    sN[i * 4 + 1] = (t4.y == 1 ? BIGV : 0.0f) + i2;
    sN[i * 4 + 2] = (t4.z == 1 ? BIGV : 0.0f) + i2;
    sN[i * 4 + 3] = (t4.w == 1 ? BIGV : 0.0f) + i2;
    __syncthreads();

    float mp0 = 3.0e38f, mp1 = 3.0e38f, mp2 = 3.0e38f, mp3 = 3.0e38f;
    float mn0 = 3.0e38f, mn1 = 3.0e38f, mn2 = 3.0e38f, mn3 = 3.0e38f;
    const float m2i = -2.0f * (float)i;
    #pragma unroll 4
    for (int j = 0; j < HH; ++j) {
        const float t2 = m2i * (float)j;                 // exact integer < 2^24
        const float4 p = *(const float4*)&sP[j * 4];
        const float4 n = *(const float4*)&sN[j * 4];
        mp0 = fminf(mp0, p.x + t2); mp1 = fminf(mp1, p.y + t2);
        mp2 = fminf(mp2, p.z + t2); mp3 = fminf(mp3, p.w + t2);
        mn0 = fminf(mn0, n.x + t2); mn1 = fminf(mn1, n.y + t2);
        mn2 = fminf(mn2, n.z + t2); mn3 = fminf(mn3, n.w + t2);
    }
    const int ob = (b * HH + i) * WW + w0;
    float4 op = { mp0 + i2, mp1 + i2, mp2 + i2, mp3 + i2 };
    float4 on = { mn0 + i2, mn1 + i2, mn2 + i2, mn3 + i2 };
    *(float4*)(d2p + ob) = op;
    *(float4*)(d2n + ob) = on;
}

// Pass 2: envelope along W + full fusion (sqrt, sigmoid, weighted sums).
// One block = one (b,h) row. The LDS staging is a pure byte-copy of the raw
// d2 rows (the +j^2 fold moved into the inner loop as an exact integer fma:
// cost(j) = j^2 - 2*w*j), so it rides GLOBAL_LOAD_ASYNC_TO_LDS_B32 and the
// ASYNCcnt counter, overlapping with the sigmoid computation. WMMA reductions.
__global__ void edt_w_fuse(const float* __restrict__ d2ph, const float* __restrict__ d2nh,
                           const int* __restrict__ tg, const float* __restrict__ logits,
                           float* __restrict__ part) {
    __shared__ __align__(16) float sP[WW];
    __shared__ __align__(16) float sN[WW];
    __shared__ float redP[8], redW[8], redC[8];
    const int b = blockIdx.x >> 8;
    const int h = blockIdx.x & 255;
    const int w = threadIdx.x;
    const int base = (b * HH + h) * WW;
    const float w2 = (float)(w * w);

#if USE_ASYNC_LDS
    __builtin_amdgcn_global_load_async_to_lds_b32(
        (async_gint*)(d2ph + base + w), (async_lint*)&sP[w], 0, 0);
    __builtin_amdgcn_global_load_async_to_lds_b32(
        (async_gint*)(d2nh + base + w), (async_lint*)&sN[w], 0, 0);
#else
    sP[w] = d2ph[base + w];
    sN[w] = d2nh[base + w];
#endif
    const int   t  = tg[base + w];
    const float l0 = logits[((b * 2 + 0) * HH + h) * WW + w];
    const float l1 = logits[((b * 2 + 1) * HH + h) * WW + w];
    const float pred = 1.0f / (1.0f + expf(l0 - l1));   // softmax(ch=2)[:,1]
#if USE_ASYNC_LDS
    __builtin_amdgcn_s_wait_asynccnt(0);
#endif
    __syncthreads();

    float mp = 3.0e38f, mn = 3.0e38f;
    const float m2w = -2.0f * (float)w;
    for (int j = 0; j < WW; j += 4) {
        const float4 p = *(const float4*)&sP[j];
        const float4 n = *(const float4*)&sN[j];
        const float j0 = (float)(j),      j1 = (float)(j + 1);
        const float j2f = (float)(j + 2), j3 = (float)(j + 3);
        const float t0 = fmaf(j0, j0, m2w * j0);   // j^2 - 2wj, exact < 2^24
        const float t1 = fmaf(j1, j1, m2w * j1);
        const float t2 = fmaf(j2f, j2f, m2w * j2f);
        const float t3 = fmaf(j3, j3, m2w * j3);
        mp = fminf(mp, p.x + t0); mn = fminf(mn, n.x + t0);
        mp = fminf(mp, p.y + t1); mn = fminf(mn, n.y + t1);
        mp = fminf(mp, p.z + t2); mn = fminf(mn, n.z + t2);
        mp = fminf(mp, p.w + t3); mn = fminf(mn, n.w + t3);
    }
    const float d2pv = mp + w2;
    const float d2nv = mn + w2;
    const float sdt  = sqrtf(d2pv) - sqrtf(d2nv);
    const float wsum = pred * sdt;
    const float cf   = (t == 1) ? 1.0f : 0.0f;

    // tensor-op wave reductions (EXEC all-ones here: no divergence yet)
    const float rp = wave_reduce32(pred);
    const float rw = wave_reduce32(wsum);
    const float rc = wave_reduce32(cf);

    const int wave = threadIdx.x >> 5;
    const int lane = threadIdx.x & 31;
    if (lane == 0) { redP[wave] = rp; redW[wave] = rw; redC[wave] = rc; }
    __syncthreads();
    if (threadIdx.x == 0) {
        float P = 0.0f, Wt = 0.0f, Cc = 0.0f;
        #pragma unroll
        for (int k = 0; k < 8; ++k) { P += redP[k]; Wt += redW[k]; Cc += redC[k]; }
        float* o = part + 3 * blockIdx.x;
        o[0] = P; o[1] = Wt; o[2] = Cc;
    }
}

// Pass 3: per-image guards and final mean (fixed order -> deterministic).
__global__ void finalize(const float* __restrict__ part, float* __restrict__ out) {
    __shared__ float perb[BB];
    const int t = threadIdx.x;
    if (t < BB) {
        float P = 0.0f, Wt = 0.0f, Cc = 0.0f;
        const float* p = part + 3 * (t * HH);
        for (int l = 0; l < HH; ++l) {
            P += p[3 * l + 0]; Wt += p[3 * l + 1]; Cc += p[3 * l + 2];
        }
        const float inv = 1.0f / (float)(HH * WW);
        const float meanPred = P * inv;
        const float meanW    = Wt * inv;
        float pb;
        if (Cc == 0.0f)                   pb = meanPred;
        else if (Cc == (float)(HH * WW))  pb = 1.0f - meanPred;
        else                              pb = meanW;
        perb[t] = pb;
    }
    __syncthreads();
    if (t == 0) {
        float s = 0.0f;
        #pragma unroll
        for (int k = 0; k < BB; ++k) s += perb[k];
        out[0] = s * (1.0f / (float)BB);
    }
}

extern "C" void kernel_launch(void* const* d_in, const int* in_sizes, int n_in,
                              void* d_out, int out_size, void* d_ws, size_t ws_size,
                              hipStream_t stream) {
    (void)in_sizes; (void)n_in; (void)out_size; (void)ws_size;
    const float* logits  = (const float*)d_in[0];   // [8,2,256,256] f32
    const int*   targets = (const int*)d_in[1];     // [8,256,256] i32

    float* d2p  = (float*)d_ws;                         // 2 MB
    float* d2n  = d2p + (size_t)BB * HH * WW;           // 2 MB
    float* part = d2n + (size_t)BB * HH * WW;           // 24 KB

    edt_h_pass<<<BB * (WW / 4), 256, 0, stream>>>(targets, d2p, d2n);
    edt_w_fuse<<<BB * HH, 256, 0, stream>>>(d2p, d2n, targets, logits, part);
    finalize<<<1, 32, 0, stream>>>(part, (float*)d_out);
}